// HOG_55216099557720
// MI455X (gfx1250) — compile-verified
//
#include <hip/hip_runtime.h>
#include <stdint.h>

// ---------------------------------------------------------------------------
// HOG on MI455X (gfx1250).
// Bandwidth/VALU-bound (50 MB @ 23.3 TB/s => ~2.3us floor; ~1 Gop irregular
// VALU). No matmul structure => no WMMA; the CDNA5 paths that pay are the
// async global->LDS copies (ASYNCcnt, B128 for the aligned interior), wave32
// DS vector loads, and cross-lane reduction. This revision removes the
// div/mod staging address math, shares gradient subexpressions across the
// 8-pixel run, and uses v_rcp/v_sqrt instead of full-precision div/sqrt.
// ---------------------------------------------------------------------------

namespace {
constexpr int BATCH = 16;
constexpr int CH    = 3;
constexpr int IMH   = 512;
constexpr int IMW   = 512;
constexpr int CELL  = 16;
constexpr int HC    = IMH / CELL;   // 32 cell rows
constexpr int WC    = IMW / CELL;   // 32 cell cols
constexpr int NB    = 9;
constexpr int HB    = HC - 1;       // 31
constexpr int WB    = WC - 1;       // 31

// Kernel A tiling: one workgroup = one row of 8 cells (128x16 px) + halo.
constexpr int TCX    = 8;
constexpr int TPX    = TCX * CELL;        // 128 px wide
constexpr int LROWS  = CELL + 2;          // 18 rows (1 halo each side)
constexpr int NROWS  = CH * LROWS;        // 54 staged rows
constexpr int LPITCH = 136;               // pitch in words
// LDS column layout: image col (x0-1+cc) lives at LDS col cc+3, cc=0..129.
//   main  cc=1..128  -> cols 4..131 (16B aligned, b128 path, never reflected)
//   halo  cc=0,129   -> cols 3,132  (b32 path, reflected)

constexpr float PI_F   = 3.14159265358979323846f;
constexpr float HPI_F  = 1.57079632679489661923f;
constexpr float GAIN   = 2.86478897565412f;    // 9 / pi
} // namespace

__device__ __forceinline__ int reflect512(int i) {
  i = (i < 0) ? -i : i;                 // reflect (no edge repeat), pad = 1
  return (i > IMH - 1) ? (2 * (IMH - 1) - i) : i;
}

// CDNA5 async global -> LDS copies (tracked by ASYNCcnt).
__device__ __forceinline__ void async_g2l_b128(unsigned lds_off, uint64_t gaddr) {
  asm volatile("global_load_async_to_lds_b128 %0, %1, off"
               :: "v"(lds_off), "v"(gaddr)
               : "memory");
}
__device__ __forceinline__ void async_g2l_b32(unsigned lds_off, uint64_t gaddr) {
  asm volatile("global_load_async_to_lds_b32 %0, %1, off"
               :: "v"(lds_off), "v"(gaddr)
               : "memory");
}
__device__ __forceinline__ void wait_async0() {
  asm volatile("s_wait_asynccnt 0" ::: "memory");
}

// atan2(y, x) mod pi, mapped to bin position bp = phase * 9/pi in [0, 9].
// Octant fold + degree-11 odd minimax polynomial (|err| ~2e-6 on [0,1]).
// phase==pi -> bp==9 -> triangle slot 9, folded into bin 0 (mod-pi wrap).
__device__ __forceinline__ float binpos(float gx, float gy) {
  const float ax = fabsf(gx), ay = fabsf(gy);
  const float mx = fmaxf(fmaxf(ax, ay), 1e-30f);
  const float mn = fminf(ax, ay);
  const float r  = mn * __builtin_amdgcn_rcpf(mx);   // v_rcp_f32, ~1ulp
  const float s  = r * r;
  float t = -0.01172120f;
  t = fmaf(t, s,  0.05265332f);
  t = fmaf(t, s, -0.11643287f);
  t = fmaf(t, s,  0.19354346f);
  t = fmaf(t, s, -0.33262347f);
  t = fmaf(t, s,  0.99997726f);
  t = t * r;                                      // atan(r) in [0, pi/4]
  if (ay > ax) t = HPI_F - t;                     // -> [0, pi/2]
  const bool signdiff = ((__float_as_uint(gx) ^ __float_as_uint(gy)) >> 31) != 0u;
  if (signdiff) t = PI_F - t;                     // -> [0, pi]
  return t * GAIN;                                // [0, 9]
}

// ---------------------------------------------------------------------------
// Kernel A: per-cell 9-bin histograms -> hist[B][HC][WC][NB]
// ---------------------------------------------------------------------------
__global__ __launch_bounds__(256) void hog_cell_hist(const float* __restrict__ im,
                                                     float* __restrict__ hist) {
  __shared__ __align__(16) float sT[NROWS * LPITCH];  // 28.7 KB

  const int tx = blockIdx.x;   // tile col (0..3)
  const int cy = blockIdx.y;   // cell row (0..31)
  const int b  = blockIdx.z;   // batch
  const int t  = threadIdx.x;
  const int x0 = tx * TPX;
  const int y0 = cy * CELL;

  // Low 32 bits of a flat LDS pointer are the LDS byte offset (ISA 10.2).
  const unsigned ldsBase = (unsigned)(uintptr_t)(&sT[0]);
  const float* imb = im + (size_t)b * CH * IMH * IMW;

  // ---- Main interior: 54 rows x 128 cols, b128 per lane, shift-only math.
  {
    const int lane4 = (t & 31) * 4;            // col within main run
#pragma unroll
    for (int i = 0; i < 7; ++i) {
      const int R = i * 8 + (t >> 5);          // staged row 0..55
      if (R < NROWS) {
        const int ch = R / LROWS;              // small-range const div
        const int rr = R - ch * LROWS;
        const int gy = reflect512(y0 + rr - 1);
        const float* src = imb + ((size_t)(ch * IMH + gy) * IMW + x0 + lane4);
        const unsigned dst = ldsBase + (unsigned)((R * LPITCH + 4 + lane4) * 4);
        async_g2l_b128(dst, (uint64_t)src);
      }
    }
  }
  // ---- Halo: 2 reflected columns per staged row (108 b32 loads).
  if (t < 2 * NROWS) {
    const int side = t & 1;                    // 0 = left, 1 = right
    const int R    = t >> 1;
    const int ch   = R / LROWS;
    const int rr   = R - ch * LROWS;
    const int gy   = reflect512(y0 + rr - 1);
    const int gx   = reflect512(side ? (x0 + TPX) : (x0 - 1));
    const float* src = imb + ((size_t)(ch * IMH + gy) * IMW + gx);
    const unsigned dst = ldsBase + (unsigned)((R * LPITCH + 3 + side * (TPX + 1)) * 4);
    async_g2l_b32(dst, (uint64_t)src);
  }
  wait_async0();     // each wave drains its own ASYNCcnt
  __syncthreads();   // then cross-wave visibility

  // ---- Each wave owns one cell; each lane owns (row, 8-px segment) ----
  const int wave = t >> 5;            // cell within tile (0..7)
  const int lane = t & 31;
  const int prow = lane >> 1;         // 0..15 pixel row in cell
  const int seg  = lane & 1;
  const int px0  = wave * CELL + seg * 8;   // tile col of first pixel (mult of 8)

  float bgx[8], bgy[8], bm2[8];
#pragma unroll
  for (int i = 0; i < 8; ++i) { bgx[i] = 0.f; bgy[i] = 0.f; bm2[i] = -1.f; }

#pragma unroll
  for (int ch = 0; ch < CH; ++ch) {
    float w[3][10];   // window idx j <-> LDS col px0+3+j
#pragma unroll
    for (int r = 0; r < 3; ++r) {
      const int R = ch * LROWS + prow + r;
      w[r][0] = sT[R * LPITCH + px0 + 3];                      // ds_load_b32
      const float4 a = *(const float4*)&sT[R * LPITCH + px0 + 4];  // b128
      const float4 c = *(const float4*)&sT[R * LPITCH + px0 + 8];  // b128
      w[r][1]=a.x; w[r][2]=a.y; w[r][3]=a.z; w[r][4]=a.w;
      w[r][5]=c.x; w[r][6]=c.y; w[r][7]=c.z; w[r][8]=c.w;
      w[r][9] = sT[R * LPITCH + px0 + 12];                     // ds_load_b32
    }
    // Shared subexpressions across the 8-pixel run.
    float s02[10], rd[10];
#pragma unroll
    for (int j = 0; j < 10; ++j) {
      s02[j] = w[0][j] + w[2][j];      // for gx: top+bottom rows
      rd[j]  = w[2][j] - w[0][j];      // for gy: bottom-top rows
    }
#pragma unroll
    for (int i = 0; i < 8; ++i) {
      const float gx = fmaf(2.f, w[1][i+2] - w[1][i], s02[i+2] - s02[i]);
      const float gy = fmaf(2.f, rd[i+1], rd[i] + rd[i+2]);
      const float m2 = fmaf(gx, gx, gy * gy);
      if (m2 > bm2[i]) { bm2[i] = m2; bgx[i] = gx; bgy[i] = gy; }  // first-max
    }
  }

  // ---- Orientation binning: triangle kernel over 10 slots (9 wraps -> 0).
  // Bilinear soft binning == slot k receives max(0, 1-|bp-k|) * mag.
  float h[NB + 1];
#pragma unroll
  for (int k = 0; k <= NB; ++k) h[k] = 0.f;

#pragma unroll
  for (int i = 0; i < 8; ++i) {
    const float mag = __builtin_amdgcn_sqrtf(bm2[i]);   // v_sqrt_f32
    const float bp  = binpos(bgx[i], bgy[i]);           // [0, 9]
#pragma unroll
    for (int k = 0; k <= NB; ++k) {
      const float wgt = fmaxf(1.f - fabsf(bp - (float)k), 0.f);
      h[k] = fmaf(wgt, mag, h[k]);
    }
  }
  h[0] += h[NB];                         // ri = (left+1) % 9  /  phase==pi wrap

  // ---- Wave32 butterfly reduction: 32 lanes cover the cell's 256 px ----
#pragma unroll
  for (int k = 0; k < NB; ++k) {
    float v = h[k];
    v += __shfl_xor(v, 16, 32);
    v += __shfl_xor(v,  8, 32);
    v += __shfl_xor(v,  4, 32);
    v += __shfl_xor(v,  2, 32);
    v += __shfl_xor(v,  1, 32);
    h[k] = v;
  }
  if (lane == 0) {
    const int cell = (b * HC + cy) * WC + tx * TCX + wave;
#pragma unroll
    for (int k = 0; k < NB; ++k)
      hist[cell * NB + k] = h[k] * (1.f / 256.f);
  }
}

// ---------------------------------------------------------------------------
// Kernel B: overlapping 2x2 block L2-Hys normalization.
// out[b][ib][jb][di][dj][k] over 36-value groups; hist fits in L2.
// ---------------------------------------------------------------------------
__global__ __launch_bounds__(256) void hog_block_norm(const float* __restrict__ hist,
                                                      float* __restrict__ out) {
  const int tid = blockIdx.x * blockDim.x + threadIdx.x;
  if (tid >= BATCH * HB * WB) return;
  const int b  = tid / (HB * WB);
  const int r  = tid - b * (HB * WB);
  const int ib = r / WB;
  const int jb = r - ib * WB;

  float v[4 * NB];
#pragma unroll
  for (int di = 0; di < 2; ++di)
#pragma unroll
    for (int dj = 0; dj < 2; ++dj) {
      const float* src = hist + ((b * HC + ib + di) * WC + (jb + dj)) * NB;
#pragma unroll
      for (int k = 0; k < NB; ++k)
        v[(di * 2 + dj) * NB + k] = src[k];
    }

  float s1 = 0.f;
#pragma unroll
  for (int k = 0; k < 36; ++k) s1 += v[k] * v[k];
  const float inv1 = 1.f / (sqrtf(s1) + 1e-10f);

  float s2 = 0.f;
#pragma unroll
  for (int k = 0; k < 36; ++k) {
    float tv = v[k] * inv1;
    tv = tv > 0.2f ? 0.2f : tv;
    v[k] = tv;
    s2 += tv * tv;
  }
  const float inv2 = 1.f / (sqrtf(s2) + 1e-10f);

  float* dst = out + (size_t)tid * 36;
#pragma unroll
  for (int k = 0; k < 36; ++k) dst[k] = v[k] * inv2;
}

// ---------------------------------------------------------------------------
extern "C" void kernel_launch(void* const* d_in, const int* in_sizes, int n_in,
                              void* d_out, int out_size, void* d_ws, size_t ws_size,
                              hipStream_t stream) {
  (void)in_sizes; (void)n_in; (void)out_size; (void)ws_size;
  const float* im   = (const float*)d_in[0];
  float*       hist = (float*)d_ws;          // 16*32*32*9 fp32 = 589,824 B
  float*       out  = (float*)d_out;         // 553,536 fp32

  dim3 gridA(IMW / TPX, HC, BATCH);          // (4, 32, 16)
  hog_cell_hist<<<gridA, 256, 0, stream>>>(im, hist);

  const int nBlk = BATCH * HB * WB;          // 15,376
  hog_block_norm<<<(nBlk + 255) / 256, 256, 0, stream>>>(hist, out);
}